// OnlineHardExampleMining_28071906246689
// MI455X (gfx1250) — compile-verified
//
#include <hip/hip_runtime.h>

// OHEM loss: logits[8,19,512,512] f32 + target[8,512,512] i32 -> scalar f32.
// Bandwidth-bound: 159MB logits @ 23.3TB/s ~= 7us floor. No matmul structure -> WMMA
// does not apply. CDNA5 path used instead: Tensor Data Mover (tensor_load_to_lds)
// stages each block's 19x1024 logits tile (rows strided by HW) into LDS
// asynchronously (TENSORcnt), overlapping DMA with target loads / histogram setup.
// Top-k (num_keep) is resolved with an exact, deterministic 3-level radix-select
// (12+12+8 key bits) instead of sorting 2M floats.

#define C_    19
#define HW_   (512 * 512)
#define N_    (8 * HW_)        // 2,097,152 pixels
#define TPB   256
#define PPB   1024             // pixels per block in main pass (4 per thread)
#define NBLK1 (N_ / PPB)       // 2048 main-pass blocks
#define NBLK  (N_ / TPB)       // 8192 blocks for 1-pixel/thread passes

#if defined(__has_builtin)
#  if __has_builtin(__builtin_amdgcn_tensor_load_to_lds) && __has_builtin(__builtin_amdgcn_s_wait_tensorcnt)
#    define HAVE_TDM 1
#  endif
#endif
#ifndef HAVE_TDM
#  define HAVE_TDM 0
#endif

#if HAVE_TDM
typedef unsigned int u32x4 __attribute__((ext_vector_type(4)));
typedef int          i32x4 __attribute__((ext_vector_type(4)));
typedef int          i32x8 __attribute__((ext_vector_type(8)));
#endif

__device__ __forceinline__ unsigned fkey(float f) {
  unsigned u = __float_as_uint(f);
  return (u & 0x80000000u) ? ~u : (u | 0x80000000u);   // order-preserving float->uint
}
__device__ __forceinline__ float keyf(unsigned k) {
  unsigned u = (k & 0x80000000u) ? (k & 0x7fffffffu) : ~k;
  return __uint_as_float(u);
}

__global__ void k_zero(unsigned* __restrict__ p, int n) {
  int i = blockIdx.x * blockDim.x + threadIdx.x;
  if (i < n) p[i] = 0u;
}

// Pass 1: TDM-staged CE + validity + 4096-bin histogram of key[31:20].
__global__ __launch_bounds__(TPB) void k_loss_hist(const float* __restrict__ logits,
                                                   const int* __restrict__ target,
                                                   float* __restrict__ losses,
                                                   unsigned* __restrict__ hist1) {
  __shared__ unsigned lh[4096];          // 16 KB
#if HAVE_TDM
  __shared__ float tile[C_ * PPB];       // 76 KB: tile[c*PPB + pix]
#endif
  for (int i = threadIdx.x; i < 4096; i += TPB) lh[i] = 0u;

  int base = blockIdx.x * PPB;           // PPB divides HW_: block never crosses batch
  int b    = base >> 18;
  int hw0  = base & (HW_ - 1);
  const float* gbase = logits + (size_t)b * ((size_t)C_ * HW_) + hw0;

  // overlap target loads with the tensor DMA
  int tgt[4];
#pragma unroll
  for (int j = 0; j < 4; ++j) tgt[j] = target[base + threadIdx.x + j * TPB];

#if HAVE_TDM
  if (threadIdx.x < 32) {                // one TDM issue per block (wave 0)
    // Generic LDS pointers carry the group offset in addr[31:0] (ISA 10.2 aperture map).
    unsigned           lds_addr = (unsigned)(size_t)(void*)tile;
    unsigned long long ga       = (unsigned long long)(size_t)(const void*)gbase;

    // D# group0: count=1 (valid), lds_addr, 57b global byte address, type=2 ("image")
    u32x4 g0 = { 1u, lds_addr, (unsigned)(ga & 0xffffffffu),
                 (unsigned)(ga >> 32) | 0x80000000u };
    // D# group1: data_size=4B; tensor_dim0=HW_, tensor_dim1=C_;
    //            tile_dim0=PPB, tile_dim1=C_, tile_dim2=0; dim0_stride=HW_.
    i32x8 g1 = { (int)(2u << 16),                                     // data_size=2 (4B)
                 (int)((HW_ & 0xffffu) << 16),                        // tensor_dim0 lo16
                 (int)(((unsigned)C_ << 16) | ((HW_ >> 16) & 0xffffu)),// dim0 hi16 | dim1 lo16
                 (int)((unsigned)PPB << 16),                          // dim1 hi16=0 | tile_dim0
                 (int)C_,                                             // tile_dim1 | tile_dim2=0
                 (int)HW_,                                            // dim0_stride lo32
                 0, 0 };                                              // dim0_stride hi | dim1_stride
    i32x4 gz = { 0, 0, 0, 0 };                                        // 2D tensor: groups 2/3 unused
#if __clang_major__ >= 23
    i32x8 gz8 = { 0, 0, 0, 0, 0, 0, 0, 0 };
    __builtin_amdgcn_tensor_load_to_lds(g0, g1, gz, gz, gz8, 0);
#else
    __builtin_amdgcn_tensor_load_to_lds(g0, g1, gz, gz, 0);
#endif
    __builtin_amdgcn_s_wait_tensorcnt(0);                             // issuer waits; others cnt==0
  }
#endif
  __syncthreads();                       // hist zeroed + tile resident

#pragma unroll
  for (int j = 0; j < 4; ++j) {
    int pix = threadIdx.x + j * TPB;
    int p   = base + pix;
    int t   = tgt[j];

    float v[C_];
#if HAVE_TDM
#pragma unroll
    for (int c = 0; c < C_; ++c) v[c] = tile[c * PPB + pix];
#else
#pragma unroll
    for (int c = 0; c < C_; ++c) v[c] = gbase[(size_t)c * HW_ + pix];
#endif
    float m = v[0];
#pragma unroll
    for (int c = 1; c < C_; ++c) m = fmaxf(m, v[c]);
    float s = 0.f;
#pragma unroll
    for (int c = 0; c < C_; ++c) s += __expf(v[c] - m);
    float lse = m + __logf(s);
    float xt = v[0];
#pragma unroll
    for (int c = 1; c < C_; ++c) xt = (c == t) ? v[c] : xt;

    float loss  = lse - xt;                                  // >= 0
    bool  valid = (m - lse) < -0.35667494393873245f;         // max_prob < 0.7
    float fl    = valid ? loss : -1e30f;
    losses[p] = fl;
    atomicAdd(&lh[fkey(fl) >> 20], 1u);
  }

  __syncthreads();
  for (int i = threadIdx.x; i < 4096; i += TPB) {
    unsigned c = lh[i];
    if (c) atomicAdd(&hist1[i], c);
  }
}

// Single-thread scans: locate the bin holding the k-th largest at each radix level.
__global__ void k_scan(const unsigned* __restrict__ hist1,
                       const unsigned* __restrict__ hist2,
                       const unsigned* __restrict__ hist3,
                       unsigned* __restrict__ S, int mode) {
  if (mode == 0) {
    unsigned negbin = fkey(-1e30f) >> 20;     // all invalid pixels land exactly here
    int nv = N_ - (int)hist1[negbin];         // valid CE losses are >=0: disjoint bins
    int keep = (int)(0.7f * (float)nv);
    if (keep < 100) keep = 100;
    if (keep > nv)  keep = nv;
    S[0] = (unsigned)nv;
    S[1] = (unsigned)keep;
    unsigned bin = 0, above = 0;
    if (keep > 0) {
      unsigned acc = 0;
      for (int i = 4095; i >= 0; --i) {
        unsigned c = hist1[i];
        if (acc + c >= (unsigned)keep) { bin = (unsigned)i; above = acc; break; }
        acc += c;
      }
    }
    S[2] = bin; S[3] = above;
  } else if (mode == 1) {
    unsigned keep = S[1], rem = keep - S[3];
    unsigned bin = 0, above = 0;
    if (keep > 0) {
      unsigned acc = 0;
      for (int i = 4095; i >= 0; --i) {
        unsigned c = hist2[i];
        if (acc + c >= rem) { bin = (unsigned)i; above = acc; break; }
        acc += c;
      }
    }
    S[4] = bin; S[5] = above;
  } else {
    unsigned keep = S[1], rem = keep - S[3] - S[5];
    unsigned bin = 0, above = 0;
    if (keep > 0) {
      unsigned acc = 0;
      for (int i = 255; i >= 0; --i) {
        unsigned c = hist3[i];
        if (acc + c >= rem) { bin = (unsigned)i; above = acc; break; }
        acc += c;
      }
    }
    S[6] = bin; S[7] = above;
    S[8] = (S[2] << 20) | (S[4] << 8) | bin;  // exact k-th key
    S[9] = S[3] + S[5] + above;               // count strictly greater than k-th
  }
}

// Radix level 2: elements with key[31:20]==bin1, histogram key[19:8].
__global__ __launch_bounds__(TPB) void k_filter_mid(const float* __restrict__ losses,
                                                    const unsigned* __restrict__ S,
                                                    unsigned* __restrict__ hist2) {
  __shared__ unsigned lh[4096];
  for (int i = threadIdx.x; i < 4096; i += TPB) lh[i] = 0u;
  __syncthreads();
  unsigned bin1 = S[2];
  int p = blockIdx.x * TPB + threadIdx.x;
  unsigned k = fkey(losses[p]);
  if ((k >> 20) == bin1) atomicAdd(&lh[(k >> 8) & 4095u], 1u);
  __syncthreads();
  for (int i = threadIdx.x; i < 4096; i += TPB) {
    unsigned c = lh[i];
    if (c) atomicAdd(&hist2[i], c);
  }
}

// Radix level 3: elements with key[31:8]==(bin1<<12)|bin2, histogram key[7:0].
__global__ __launch_bounds__(TPB) void k_filter_low(const float* __restrict__ losses,
                                                    const unsigned* __restrict__ S,
                                                    unsigned* __restrict__ hist3) {
  __shared__ unsigned lh[256];
  if (threadIdx.x < 256) lh[threadIdx.x] = 0u;
  __syncthreads();
  unsigned pre = (S[2] << 12) | S[4];
  int p = blockIdx.x * TPB + threadIdx.x;
  unsigned k = fkey(losses[p]);
  if ((k >> 8) == pre) atomicAdd(&lh[k & 255u], 1u);
  __syncthreads();
  if (threadIdx.x < 256) {
    unsigned c = lh[threadIdx.x];
    if (c) atomicAdd(&hist3[threadIdx.x], c);
  }
}

// Deterministic blocked sum of losses with key strictly greater than the k-th key.
__global__ __launch_bounds__(TPB) void k_sum(const float* __restrict__ losses,
                                             const unsigned* __restrict__ S,
                                             float* __restrict__ partials) {
  __shared__ float red[TPB];
  unsigned kth = S[8];
  int p = blockIdx.x * TPB + threadIdx.x;
  float v = losses[p];
  red[threadIdx.x] = (fkey(v) > kth) ? v : 0.f;
  __syncthreads();
  for (int s = TPB / 2; s > 0; s >>= 1) {
    if (threadIdx.x < s) red[threadIdx.x] += red[threadIdx.x + s];
    __syncthreads();
  }
  if (threadIdx.x == 0) partials[blockIdx.x] = red[0];
}

__global__ __launch_bounds__(TPB) void k_final(const float* __restrict__ partials,
                                               const unsigned* __restrict__ S,
                                               float* __restrict__ out) {
  __shared__ float red[TPB];
  float s = 0.f;
  for (int i = threadIdx.x; i < NBLK; i += TPB) s += partials[i];  // fixed order
  red[threadIdx.x] = s;
  __syncthreads();
  for (int t = TPB / 2; t > 0; t >>= 1) {
    if (threadIdx.x < t) red[threadIdx.x] += red[threadIdx.x + t];
    __syncthreads();
  }
  if (threadIdx.x == 0) {
    unsigned nv = S[0];
    int   keep  = (int)S[1];
    int   cgt   = (int)S[9];
    float kthv  = keyf(S[8]);
    float total = red[0] + (float)(keep - cgt) * kthv;   // exact tie handling
    float mean  = total / (float)(keep > 0 ? keep : 1);
    out[0] = (nv > 0) ? mean : 0.f;
  }
}

extern "C" void kernel_launch(void* const* d_in, const int* in_sizes, int n_in,
                              void* d_out, int out_size, void* d_ws, size_t ws_size,
                              hipStream_t stream) {
  const float* logits = (const float*)d_in[0];   // [8,19,512,512] f32
  const int*   target = (const int*)d_in[1];     // [8,512,512] i32
  float*       out    = (float*)d_out;           // scalar

  // workspace layout (~8.42 MB)
  char*     ws       = (char*)d_ws;
  float*    losses   = (float*)ws;                                 // N_ floats
  unsigned* hist1    = (unsigned*)(ws + (size_t)N_ * sizeof(float));
  unsigned* hist2    = hist1 + 4096;
  unsigned* hist3    = hist2 + 4096;
  unsigned* S        = hist3 + 256;                                // 16 scalars
  float*    partials = (float*)(S + 16);                           // NBLK floats

  const int nz = 4096 + 4096 + 256 + 16;   // re-zero every call (graph replays)
  k_zero<<<(nz + TPB - 1) / TPB, TPB, 0, stream>>>(hist1, nz);

  k_loss_hist <<<NBLK1, TPB, 0, stream>>>(logits, target, losses, hist1);
  k_scan      <<<1, 1, 0, stream>>>(hist1, hist2, hist3, S, 0);
  k_filter_mid<<<NBLK, TPB, 0, stream>>>(losses, S, hist2);
  k_scan      <<<1, 1, 0, stream>>>(hist1, hist2, hist3, S, 1);
  k_filter_low<<<NBLK, TPB, 0, stream>>>(losses, S, hist3);
  k_scan      <<<1, 1, 0, stream>>>(hist1, hist2, hist3, S, 2);
  k_sum       <<<NBLK, TPB, 0, stream>>>(losses, S, partials);
  k_final     <<<1, TPB, 0, stream>>>(partials, S, out);
}